// HardNMS_38654705664677
// MI455X (gfx1250) — compile-verified
//
#include <hip/hip_runtime.h>
#include <hip/hip_bf16.h>

// HardNMS: xp = relu(x1 - x0 - 0.1); xNMS = 256 * prod_{8 nbrs} relu(xp - nb);
// out = xp * maxpool3x3(xNMS).  Memory-bound radius-2 stencil -> LDS tiling +
// TDM (tensor_store_from_lds) async output store on gfx1250.

#define HH 1024
#define WW 1024
#define OUT_H 32
#define OUT_W 128
#define XP_H (OUT_H + 4)    // 36
#define XP_W (OUT_W + 4)    // 132
#define NMS_H (OUT_H + 2)   // 34
#define NMS_W (OUT_W + 2)   // 130
#define EPSV 0.1f
#define NTHREADS 256

typedef unsigned int v4u __attribute__((ext_vector_type(4)));
typedef int          v8i __attribute__((ext_vector_type(8)));
typedef int          v4i __attribute__((ext_vector_type(4)));

__global__ __launch_bounds__(NTHREADS)
void hardnms_tile_kernel(const float* __restrict__ x, float* __restrict__ out) {
    __shared__ float s_xp [XP_H * XP_W];    // xp with 2-halo (zero padded at image edge)
    __shared__ float s_nms[NMS_H * NMS_W];  // xNMS with 1-halo
    __shared__ float s_out[OUT_H * OUT_W];  // output tile staged for TDM store

    const int tid = threadIdx.x;
    const int w0  = blockIdx.x * OUT_W;
    const int h0  = blockIdx.y * OUT_H;
    const int b   = blockIdx.z;

    const float* __restrict__ x0p = x + (size_t)(b * 2 + 0) * HH * WW;
    const float* __restrict__ x1p = x + (size_t)(b * 2 + 1) * HH * WW;

    // ---- Stage 1: xp tile (halo 2), zero outside image (== conv zero-pad) ----
    for (int i = tid; i < XP_H * XP_W; i += NTHREADS) {
        const int ly = i / XP_W;
        const int lx = i - ly * XP_W;
        const int gh = h0 - 2 + ly;
        const int gw = w0 - 2 + lx;
        float v = 0.0f;
        if ((unsigned)gh < (unsigned)HH && (unsigned)gw < (unsigned)WW) {
            const size_t o = (size_t)gh * WW + gw;
            v = fmaxf(x1p[o] - x0p[o] - EPSV, 0.0f);
        }
        s_xp[i] = v;
    }
    __syncthreads();

    // ---- Stage 2: xNMS tile (halo 1): 256 * prod relu(center - neighbor) ----
    for (int i = tid; i < NMS_H * NMS_W; i += NTHREADS) {
        const int ly = i / NMS_W;
        const int lx = i - ly * NMS_W;
        const float c = s_xp[(ly + 1) * XP_W + (lx + 1)];
        float prod = 1.0f;
#pragma unroll
        for (int dy = 0; dy < 3; ++dy) {
#pragma unroll
            for (int dx = 0; dx < 3; ++dx) {
                if (dy == 1 && dx == 1) continue;
                const float nb = s_xp[(ly + dy) * XP_W + (lx + dx)];
                prod *= fmaxf(c - nb, 0.0f);
            }
        }
        s_nms[i] = prod * 256.0f;
    }
    __syncthreads();

    // ---- Stage 3: 3x3 max-pool of xNMS, times center xp -> out tile in LDS ----
    for (int i = tid; i < OUT_H * OUT_W; i += NTHREADS) {
        const int oy = i / OUT_W;
        const int ox = i - oy * OUT_W;
        float m = s_nms[oy * NMS_W + ox];
#pragma unroll
        for (int dy = 0; dy < 3; ++dy) {
#pragma unroll
            for (int dx = 0; dx < 3; ++dx) {
                if (dy == 0 && dx == 0) continue;
                m = fmaxf(m, s_nms[(oy + dy) * NMS_W + (ox + dx)]);
            }
        }
        s_out[i] = s_xp[(oy + 2) * XP_W + (ox + 2)] * m;
    }
    __syncthreads();

    float* __restrict__ outp = out + ((size_t)b * HH + h0) * WW + w0;

#if defined(__HIP_DEVICE_COMPILE__) && __has_builtin(__builtin_amdgcn_tensor_store_from_lds)
    // ---- Stage 4: one wave issues TDM async store of the 128x32 f32 tile ----
    if (tid < 32) {  // wave 0 (wave32); TDM ignores EXEC, issues once per wave
        const unsigned lds_off = (unsigned)(uintptr_t)(&s_out[0]);   // low 32b = LDS offset
        const unsigned long long ga =
            ((unsigned long long)(uintptr_t)outp) & 0x01FFFFFFFFFFFFFFull;  // 57-bit

        // D# group 0 (128b): count=1 | lds_addr | global_addr[56:0] | type=2
        v4u g0;
        g0.x = 1u;                                   // count=1 (valid), user mode
        g0.y = lds_off;                              // LDS byte address of tile
        g0.z = (unsigned)(ga & 0xFFFFFFFFu);         // global_addr[31:0]
        g0.w = (unsigned)((ga >> 32) & 0x01FFFFFFu)  // global_addr[56:32]
             | (2u << 30);                           // type = 2 ("image")

        // D# group 1 (256b): data_size=4B, tensor dims (remaining extent),
        // tile 128x32, row stride 1024 elements.
        const unsigned td0 = (unsigned)(WW - w0);    // remaining cols  (>= 128)
        const unsigned td1 = (unsigned)(HH - h0);    // remaining rows  (>= 32)
        v8i g1;
        g1[0] = (int)(2u << 16);                            // data_size=2 -> 4 bytes
        g1[1] = (int)((td0 & 0xFFFFu) << 16);               // tensor_dim0[15:0] @ bit48
        g1[2] = (int)(((td0 >> 16) & 0xFFFFu)
                    | ((td1 & 0xFFFFu) << 16));             // dim0 hi | dim1 lo
        g1[3] = (int)(((td1 >> 16) & 0xFFFFu)
                    | ((unsigned)OUT_W << 16));             // dim1 hi | tile_dim0=128
        g1[4] = (int)((unsigned)OUT_H);                     // tile_dim1=32, tile_dim2=0
        g1[5] = (int)((unsigned)WW);                        // tensor_dim0_stride[31:0]
        g1[6] = 0;                                          // stride hi | dim1_stride lo
        g1[7] = 0;

        const v4i z4 = {0, 0, 0, 0};                 // groups 2/3 unused -> 2D tensor
        const v8i z8 = {0, 0, 0, 0, 0, 0, 0, 0};     // trailing group (6-arg form), unused
        __builtin_amdgcn_tensor_store_from_lds(g0, g1, z4, z4, z8, 0);
        __builtin_amdgcn_s_wait_tensorcnt(0);        // LDS must stay live until done
    }
#else
    // Fallback (host pass / toolchains without the TDM builtin)
    for (int i = tid; i < OUT_H * OUT_W; i += NTHREADS) {
        const int oy = i / OUT_W;
        const int ox = i - oy * OUT_W;
        outp[(size_t)oy * WW + ox] = s_out[i];
    }
#endif
}

extern "C" void kernel_launch(void* const* d_in, const int* in_sizes, int n_in,
                              void* d_out, int out_size, void* d_ws, size_t ws_size,
                              hipStream_t stream) {
    (void)n_in; (void)out_size; (void)d_ws; (void)ws_size;
    const float* x  = (const float*)d_in[0];
    float* out      = (float*)d_out;
    const int B     = in_sizes[0] / (2 * HH * WW);   // 16
    dim3 grid(WW / OUT_W, HH / OUT_H, B);            // (8, 32, 16)
    hardnms_tile_kernel<<<grid, NTHREADS, 0, stream>>>(x, out);
}